// ExphormerAttention_17549236371612
// MI455X (gfx1250) — compile-verified
//
#include <hip/hip_runtime.h>
#include <hip/hip_bf16.h>
#include <math.h>

// ---------------------------------------------------------------------------
// Exphormer-style edge attention for MI455X (gfx1250, wave32).
// GEMMs use V_WMMA_F32_16X16X4_F32 (fp32 matrix pipe, matches reference fp32).
// The edge-feature GEMM (E = edge_attr @ We + be) is fused into the edge
// kernel and staged in LDS: saves ~614 MB of HBM round-trip for E.
// ---------------------------------------------------------------------------

typedef __attribute__((ext_vector_type(2))) float v2f;
typedef __attribute__((ext_vector_type(8))) float v8f;

#define IN_DIM   96
#define NUM_HEADS 8
#define HEAD_DIM 12
#define KSTEPS   (IN_DIM / 4)   // 24 K-steps of 4 for 16x16x4 WMMA
#define NTILES   (IN_DIM / 16)  // 6 N-tiles of 16

// --- WMMA fragment loaders (ISA 7.12.2 layouts, wave32) --------------------
// A (16x4, MxK): lanes 0-15 hold M=lane, K={0,1}; lanes 16-31 hold M=lane-16,
// K={2,3}.  Two fp32 per lane.
__device__ __forceinline__ v2f ld_afrag(const float* __restrict__ row96,
                                        int kk, int kgrp) {
  // element offset 4*kk + 2*kgrp is even -> 8B aligned
  return *(const v2f*)(row96 + 4 * kk + 2 * kgrp);
}

// B (4x16, KxN): lanes 0-15 hold N=lane with K={0,1}; lanes 16-31 hold
// N=lane-16 with K={2,3}.  W is row-major [96 in][96 out], so B[k][n] = W[k][n].
__device__ __forceinline__ v2f ld_bfrag(const float* __restrict__ W,
                                        int kk, int nn, int lane) {
  const int n = 16 * nn + (lane & 15);
  const int k = 4 * kk + 2 * (lane >> 4);
  v2f b;
  b.x = W[(size_t)k * IN_DIM + n];
  b.y = W[(size_t)(k + 1) * IN_DIM + n];
  return b;
}

// C/D (16x16 fp32): reg r holds row (r + 8*(lane>>4)), col = lane&15.

// ---------------------------------------------------------------------------
// Kernel 1: fused Q/K/V projection.  One wave per 16-row node tile.
// ---------------------------------------------------------------------------
__global__ __launch_bounds__(256) void qkv_wmma_kernel(
    const float* __restrict__ x,
    const float* __restrict__ Wq, const float* __restrict__ bq,
    const float* __restrict__ Wk, const float* __restrict__ bk,
    const float* __restrict__ Wv, const float* __restrict__ bv,
    float* __restrict__ Q, float* __restrict__ K, float* __restrict__ V,
    int n_nodes) {
  const int wave = threadIdx.x >> 5;
  const int lane = threadIdx.x & 31;
  const int tile = blockIdx.x * (blockDim.x >> 5) + wave;
  const int m0   = tile * 16;
  if (m0 >= n_nodes) return;                 // wave-uniform: no partial EXEC

  const int m    = lane & 15;
  const int kgrp = lane >> 4;
  const float* xrow = x + (size_t)(m0 + m) * IN_DIM;

  // Preload all A fragments for this 16x96 tile (48 fp32 per lane).
  v2f a[KSTEPS];
#pragma unroll
  for (int kk = 0; kk < KSTEPS; ++kk) a[kk] = ld_afrag(xrow, kk, kgrp);

  const float* Ws[3] = {Wq, Wk, Wv};
  const float* bs[3] = {bq, bk, bv};
  float*       Os[3] = {Q, K, V};

  for (int mat = 0; mat < 3; ++mat) {
    const float* __restrict__ W = Ws[mat];
    const float* __restrict__ b = bs[mat];
    float* __restrict__ O = Os[mat];
    for (int nn = 0; nn < NTILES; ++nn) {
      v8f acc = {};
#pragma unroll
      for (int kk = 0; kk < KSTEPS; ++kk) {
        v2f bf = ld_bfrag(W, kk, nn, lane);
        acc = __builtin_amdgcn_wmma_f32_16x16x4_f32(
            false, a[kk], false, bf, (short)0, acc, false, false);
      }
      const int col  = 16 * nn + m;
      const float bias = b[col];
#pragma unroll
      for (int r = 0; r < 8; ++r) {
        const int row = m0 + r + kgrp * 8;
        O[(size_t)row * IN_DIM + col] = acc[r] + bias;
      }
    }
  }
}

// ---------------------------------------------------------------------------
// Kernel 2: fused edge kernel.  One wave per 16-edge tile.
//   Phase 1: E-tile = edge_attr_tile @ We + be  via WMMA -> LDS stage.
//   Phase 2: per-edge scores + atomic scatter of messages / normalizers.
// ---------------------------------------------------------------------------
#define EDGE_WAVES 4
__global__ __launch_bounds__(EDGE_WAVES * 32) void edge_attn_kernel(
    const float* __restrict__ edge_attr,
    const int*   __restrict__ edge_index,   // [2, E] (row 0 = src, row 1 = dst)
    const float* __restrict__ We, const float* __restrict__ be,
    const float* __restrict__ Q, const float* __restrict__ K,
    const float* __restrict__ V,
    float* __restrict__ wV, float* __restrict__ Z,
    int n_edges) {
  __shared__ float e_stage[EDGE_WAVES][16][IN_DIM];   // 24 KB

  const int wave = threadIdx.x >> 5;
  const int lane = threadIdx.x & 31;
  const int tile = blockIdx.x * EDGE_WAVES + wave;
  const int e0   = tile * 16;
  const bool active = (e0 < n_edges);        // uniform per wave

  const int m    = lane & 15;
  const int kgrp = lane >> 4;

  if (active) {
    const float* arow = edge_attr + (size_t)(e0 + m) * IN_DIM;
    v2f a[KSTEPS];
#pragma unroll
    for (int kk = 0; kk < KSTEPS; ++kk) a[kk] = ld_afrag(arow, kk, kgrp);

    for (int nn = 0; nn < NTILES; ++nn) {
      v8f acc = {};
#pragma unroll
      for (int kk = 0; kk < KSTEPS; ++kk) {
        v2f bf = ld_bfrag(We, kk, nn, lane);
        acc = __builtin_amdgcn_wmma_f32_16x16x4_f32(
            false, a[kk], false, bf, (short)0, acc, false, false);
      }
      const int col  = 16 * nn + m;
      const float bias = be[col];
#pragma unroll
      for (int r = 0; r < 8; ++r)
        e_stage[wave][r + kgrp * 8][col] = acc[r] + bias;
    }
  }
  __syncthreads();

  if (active) {
    // lane = head*4 + quad; each quad lane covers 3 of the 12 head dims
    const int h  = lane >> 2;
    const int q4 = lane & 3;
    const int d0 = h * HEAD_DIM + q4 * 3;
    const float inv_sqrt_d = 0.28867513459481288f;   // 1/sqrt(12)

    for (int i = 0; i < 16; ++i) {
      const int e   = e0 + i;
      const int src = edge_index[e];
      const int dst = edge_index[n_edges + e];

      const float* kp = K + (size_t)src * IN_DIM + d0;
      const float* qp = Q + (size_t)dst * IN_DIM + d0;
      const float* er = &e_stage[wave][i][d0];

      float s = 0.f;
#pragma unroll
      for (int j = 0; j < 3; ++j) s += kp[j] * qp[j] * er[j];
      s *= inv_sqrt_d;
      // reduce the 4 partials inside the quad (butterfly -> all lanes hold sum)
      s += __shfl_xor(s, 1, 4);
      s += __shfl_xor(s, 2, 4);
      s = __expf(fminf(fmaxf(s, -5.f), 5.f));

      const float* vp = V + (size_t)src * IN_DIM + d0;
      float* wp = wV + (size_t)dst * IN_DIM + d0;
#pragma unroll
      for (int j = 0; j < 3; ++j) atomicAdd(wp + j, vp[j] * s);
      if (q4 == 0) atomicAdd(Z + (size_t)dst * NUM_HEADS + h, s);
    }
  }
}

// ---------------------------------------------------------------------------
// Kernel 0: zero accumulators (d_ws is poisoned; must re-zero every launch).
// ---------------------------------------------------------------------------
__global__ void zero_kernel(float* __restrict__ p, size_t n) {
  size_t i = (size_t)blockIdx.x * blockDim.x + threadIdx.x;
  const size_t stride = (size_t)gridDim.x * blockDim.x;
  for (; i < n; i += stride) p[i] = 0.f;
}

// ---------------------------------------------------------------------------
// Kernel 3: out = wV / (Z + 1e-6)
// ---------------------------------------------------------------------------
__global__ void normalize_kernel(const float* __restrict__ wV,
                                 const float* __restrict__ Z,
                                 float* __restrict__ out, int n_nodes) {
  const int idx = blockIdx.x * blockDim.x + threadIdx.x;
  const int total = n_nodes * IN_DIM;
  if (idx >= total) return;
  const int n = idx / IN_DIM;
  const int c = idx - n * IN_DIM;
  const int h = c / HEAD_DIM;
  out[idx] = wV[idx] / (Z[(size_t)n * NUM_HEADS + h] + 1e-6f);
}

// ---------------------------------------------------------------------------
extern "C" void kernel_launch(void* const* d_in, const int* in_sizes, int n_in,
                              void* d_out, int out_size, void* d_ws, size_t ws_size,
                              hipStream_t stream) {
  const float* x         = (const float*)d_in[0];
  const int*   edge_idx  = (const int*)d_in[1];     // int32 (JAX x64 disabled)
  const float* edge_attr = (const float*)d_in[2];
  const float* Wq = (const float*)d_in[3];
  const float* bq = (const float*)d_in[4];
  const float* Wk = (const float*)d_in[5];
  const float* bk = (const float*)d_in[6];
  const float* We = (const float*)d_in[7];
  const float* be = (const float*)d_in[8];
  const float* Wv = (const float*)d_in[9];
  const float* bv = (const float*)d_in[10];
  float* out = (float*)d_out;

  const int n_nodes = in_sizes[0] / IN_DIM;
  const int n_edges = in_sizes[1] / 2;

  // Workspace layout (fp32): Q | K | V | wV | Z  = ~78.4 MB
  const size_t nfeat = (size_t)n_nodes * IN_DIM;
  float* Q  = (float*)d_ws;
  float* K  = Q + nfeat;
  float* V  = K + nfeat;
  float* wV = V + nfeat;
  float* Z  = wV + nfeat;          // n_nodes * NUM_HEADS floats

  // 0) zero accumulators (wV followed contiguously by Z)
  const size_t n_zero = nfeat + (size_t)n_nodes * NUM_HEADS;
  zero_kernel<<<4096, 256, 0, stream>>>(wV, n_zero);

  // 1) Q/K/V projection: one wave per 16-row tile, 8 waves per block
  {
    const int tiles  = (n_nodes + 15) / 16;
    const int blocks = (tiles + 7) / 8;
    qkv_wmma_kernel<<<blocks, 256, 0, stream>>>(x, Wq, bq, Wk, bk, Wv, bv,
                                                Q, K, V, n_nodes);
  }

  // 2) fused E-GEMM + attention + scatter: one wave per 16-edge tile
  {
    const int tiles  = (n_edges + 15) / 16;
    const int blocks = (tiles + EDGE_WAVES - 1) / EDGE_WAVES;
    edge_attn_kernel<<<blocks, EDGE_WAVES * 32, 0, stream>>>(
        edge_attr, edge_idx, We, be, Q, K, V, wV, Z, n_edges);
  }

  // 3) normalize
  {
    const int total  = n_nodes * IN_DIM;
    const int blocks = (total + 255) / 256;
    normalize_kernel<<<blocks, 256, 0, stream>>>(wV, Z, out, n_nodes);
  }
}